// AttentionPool_52115133169983
// MI455X (gfx1250) — compile-verified
//
#include <hip/hip_runtime.h>

typedef __attribute__((ext_vector_type(2))) float v2f;
typedef __attribute__((ext_vector_type(8))) float v8f;

#define DIM      768
#define HEADS    12
#define DH       64
#define BATCH    32
#define NTOK     4096
#define KCHUNKS  (DIM / 4)      // 192 wmma k-steps over DIM
#define SLICES   16             // token slices per batch
#define SROWS    (NTOK / SLICES)// 256 rows per workgroup
#define CHROWS   16             // rows per chunk
#define NCHUNK   (SROWS / CHROWS)
#define TSTRIDE  784            // 768 + 16 pad: rows 2 apart land 32 banks apart
#define NWAVES   16             // 512 threads
#define CT_PER_WAVE 3           // 48 column tiles / 16 waves

// One per-lane async B128: global 16B -> LDS 16B; IOFFSET advances both sides.
#define TILE_ASYNC_LOAD(OFF) \
    asm volatile("global_load_async_to_lds_b128 %0, %1, off offset:" OFF \
                 :: "v"(ldsa), "v"(ga) : "memory")

// ---------------------------------------------------------------------------
// Kernel 1: fold query through Wq, then fold q through Wk into R (768 x 16,
// heads padded 12->16 with zeros), stored pre-swizzled as WMMA B-fragments
// for V_WMMA_F32_16X16X4_F32:
//   chunk kc, lane l<16 : N=h=l,    slots = (d=4kc, 4kc+1)
//   chunk kc, lane l>=16: N=h=l-16, slots = (d=4kc+2, 4kc+3)
// Softmax scale (1/sqrt(64)) folded into R and per-head bias cb.
// ---------------------------------------------------------------------------
__global__ void __launch_bounds__(256) ap_prep(
    const float* __restrict__ query,
    const float* __restrict__ in_w,    // (2304,768): [Wq;Wk;Wv]
    const float* __restrict__ in_b,    // (2304,)
    float* __restrict__ Rfrag,         // 192*64 floats, swizzled
    float* __restrict__ cb)            // 16 floats (padded)
{
    __shared__ float q[DIM];
    const int t = threadIdx.x;

    for (int d = t; d < DIM; d += 256) {
        float acc = in_b[d];
        const float* wrow = in_w + (size_t)d * DIM;
        for (int e = 0; e < DIM; ++e) acc += query[e] * wrow[e];
        q[d] = acc;
    }
    __syncthreads();

    if (t < 16) {
        float acc = 0.f;
        if (t < HEADS)
            for (int i = 0; i < DH; ++i)
                acc += q[t * DH + i] * in_b[DIM + t * DH + i];
        cb[t] = acc * 0.125f;
    }

    for (int idx = t; idx < DIM * 16; idx += 256) {
        const int d = idx >> 4;
        const int h = idx & 15;
        float acc = 0.f;
        if (h < HEADS) {
            const float* wk = in_w + ((size_t)(DIM + h * DH)) * DIM + d;
            for (int i = 0; i < DH; ++i) acc += q[h * DH + i] * wk[(size_t)i * DIM];
            acc *= 0.125f;
        }
        const int kc   = d >> 2;
        const int lane = h + 16 * ((d >> 1) & 1);
        const int slot = d & 1;
        Rfrag[kc * 64 + lane * 2 + slot] = acc;
    }
}

// ---------------------------------------------------------------------------
// Kernel 2: fused flash-style pass. One workgroup per (batch, 256-row slice).
// Single read of the token tensor. Double-buffered ASYNC staging:
//   chunk ch+1 streams global->LDS via GLOBAL_LOAD_ASYNC_TO_LDS_B128 while
//   chunk ch runs WMMA scores (K split across 16 waves) -> reduce ->
//   online softmax -> alpha-rescale + WMMA pooling accumulate.
// Gating: s_wait_asynccnt 0 + workgroup barrier at the top of each chunk.
// ---------------------------------------------------------------------------
__global__ void __launch_bounds__(512) ap_flash(
    const float* __restrict__ tokens,
    const float* __restrict__ Rfrag_g,
    const float* __restrict__ cb_g,
    float* __restrict__ Tpart,   // (32,16,12,768)
    float* __restrict__ Mpart,   // (32,16,12)
    float* __restrict__ Spart)   // (32,16,12)
{
    __shared__ __align__(16) float ldsR[KCHUNKS * 64];          // 48 KB B-frags
    __shared__ __align__(16) float tile[2][CHROWS * TSTRIDE];   // 2x50 KB tokens
    __shared__ float sbuf[NWAVES * 256];                        // 16 KB partials
    __shared__ float sc[CHROWS * 16];                           // scores [r][h]
    __shared__ float wts[CHROWS][16];                           // weights (pad 0)
    __shared__ float alpha_s[16];
    __shared__ float m_s[16], s_s[16];
    __shared__ float cbl[16];

    const int t    = threadIdx.x;
    const int wave = t >> 5;
    const int lane = t & 31;
    const int b     = blockIdx.x >> 4;
    const int slice = blockIdx.x & 15;

    const float* tokb = tokens + ((size_t)b * NTOK + slice * SROWS) * DIM;

    // ---- async-staging geometry: 32 threads per row, 6 x 16B per thread, 512B apart ----
    const int srow  = t >> 5;          // 0..15
    const int scol4 = t & 31;          // float4 column base
    const float* grow = tokb + (size_t)srow * DIM + scol4 * 4;
    const unsigned lds0 = (unsigned)(size_t)&tile[0][srow * TSTRIDE + scol4 * 4];
    const unsigned lds1 = (unsigned)(size_t)&tile[1][srow * TSTRIDE + scol4 * 4];

    // kick off chunk 0 into buffer 0
    {
        const unsigned ldsa = lds0;
        const unsigned long long ga = (unsigned long long)(size_t)grow;
        TILE_ASYNC_LOAD("0");    TILE_ASYNC_LOAD("512");  TILE_ASYNC_LOAD("1024");
        TILE_ASYNC_LOAD("1536"); TILE_ASYNC_LOAD("2048"); TILE_ASYNC_LOAD("2560");
    }

    for (int i = t; i < (KCHUNKS * 64) / 4; i += 512)
        ((float4*)ldsR)[i] = ((const float4*)Rfrag_g)[i];
    if (t < 16) {
        m_s[t] = -1e30f; s_s[t] = 0.f; alpha_s[t] = 1.f; cbl[t] = cb_g[t];
    }

    const int mrow = lane & 15;
    const int koff = (lane >> 4) << 1;      // lanes>=16 hold K slots 2,3

    v8f acc0 = {}, acc1 = {}, acc2 = {};

    for (int ch = 0; ch < NCHUNK; ++ch) {
        const int cur = ch & 1;
        // current buffer's async loads complete, all waves' data visible
        asm volatile("s_wait_asynccnt 0x0" ::: "memory");
        __syncthreads();

        // ---- issue next chunk into the other buffer; prefetch chunk+2 ----
        if (ch + 1 < NCHUNK) {
            const unsigned ldsa = cur ? lds0 : lds1;
            const unsigned long long ga = (unsigned long long)(size_t)
                (grow + (size_t)(ch + 1) * CHROWS * DIM);
            TILE_ASYNC_LOAD("0");    TILE_ASYNC_LOAD("512");  TILE_ASYNC_LOAD("1024");
            TILE_ASYNC_LOAD("1536"); TILE_ASYNC_LOAD("2048"); TILE_ASYNC_LOAD("2560");
        }
        if (ch + 2 < NCHUNK)
            __builtin_prefetch(tokb + (size_t)(ch + 2) * CHROWS * DIM + t * 24, 0, 0);

        const float* tb = tile[cur];

        // ---- scores: K split across waves, 12 WMMA each, partial C -> LDS ----
        {
            v8f c = {};
            const float* ar = tb + mrow * TSTRIDE + koff;
            const v2f*   bp = (const v2f*)ldsR + lane;
            const int kc0 = wave * (KCHUNKS / NWAVES);
            for (int j = 0; j < KCHUNKS / NWAVES; ++j) {
                const int kc = kc0 + j;
                v2f a  = *(const v2f*)(ar + kc * 4);
                v2f bb = bp[kc * 32];
                c = __builtin_amdgcn_wmma_f32_16x16x4_f32(
                        false, a, false, bb, (short)0, c, false, false);
            }
            float* sb = sbuf + wave * 256 + lane * 8;
            for (int i = 0; i < 8; ++i) sb[i] = c[i];
        }
        __syncthreads();

        // ---- reduce partials into sc[r][h] (+bias) ----
        if (t < 256) {
            const int r = t >> 4, h = t & 15;
            if (h < HEADS) {
                const int lidx = h + ((r >> 3) << 4);   // lane holding (M=r,N=h)
                const int vi   = r & 7;                 // VGPR slot
                float s = cbl[h];
                for (int w2 = 0; w2 < NWAVES; ++w2)
                    s += sbuf[w2 * 256 + lidx * 8 + vi];
                sc[r * 16 + h] = s;
            }
        }
        __syncthreads();

        // ---- online softmax update (one thread per head) ----
        if (t < 16) {
            float al = 1.f;
            if (t < HEADS) {
                const float mold = m_s[t];
                float mc = -1e30f;
                for (int r = 0; r < CHROWS; ++r) mc = fmaxf(mc, sc[r * 16 + t]);
                const float mnew = fmaxf(mold, mc);
                al = __expf(mold - mnew);
                float ssum = 0.f;
                for (int r = 0; r < CHROWS; ++r) {
                    const float w = __expf(sc[r * 16 + t] - mnew);
                    wts[r][t] = w;
                    ssum += w;
                }
                s_s[t] = s_s[t] * al + ssum;
                m_s[t] = mnew;
            } else {
                for (int r = 0; r < CHROWS; ++r) wts[r][t] = 0.f;
            }
            alpha_s[t] = al;
        }
        __syncthreads();

        // ---- pool: rescale accumulators, 4 WMMA per owned column tile ----
        {
            float al[8];
            const int hb = (lane >> 4) << 3;            // head base for lane half
            for (int i = 0; i < 8; ++i) al[i] = alpha_s[hb + i];

            v8f* accp[CT_PER_WAVE] = { &acc0, &acc1, &acc2 };
            for (int ct = 0; ct < CT_PER_WAVE; ++ct) {
                const int col = (wave * CT_PER_WAVE + ct) * 16 + (lane & 15);
                v8f c = *accp[ct];
                for (int i = 0; i < 8; ++i) c[i] *= al[i];
                for (int kc = 0; kc < CHROWS / 4; ++kc) {
                    const int r0 = kc * 4 + koff;
                    v2f a, bb;
                    a.x  = wts[r0][lane & 15];
                    a.y  = wts[r0 + 1][lane & 15];
                    bb.x = tb[r0 * TSTRIDE + col];
                    bb.y = tb[(r0 + 1) * TSTRIDE + col];
                    c = __builtin_amdgcn_wmma_f32_16x16x4_f32(
                            false, a, false, bb, (short)0, c, false, false);
                }
                *accp[ct] = c;
            }
        }
        // loop-top asynccnt-wait + barrier doubles as the inter-chunk fence
    }
    __syncthreads();

    // ---- write per-slice partials ----
    const v8f accs[CT_PER_WAVE] = { acc0, acc1, acc2 };
    for (int ct = 0; ct < CT_PER_WAVE; ++ct) {
        const int col = (wave * CT_PER_WAVE + ct) * 16 + (lane & 15);
        for (int i = 0; i < 8; ++i) {
            const int hm = i + 8 * (lane >> 4);
            if (hm < HEADS)
                Tpart[(((size_t)b * SLICES + slice) * HEADS + hm) * DIM + col] = accs[ct][i];
        }
    }
    if (t < HEADS) {
        Mpart[((size_t)b * SLICES + slice) * HEADS + t] = m_s[t];
        Spart[((size_t)b * SLICES + slice) * HEADS + t] = s_s[t];
    }
}

// ---------------------------------------------------------------------------
// Kernel 3: merge slice partials with flash rescaling:
//   m* = max_j m_j;  denom = sum_j s_j e^{m_j-m*};
//   t[b,h,:] = sum_j e^{m_j-m*} T_j / denom
// ---------------------------------------------------------------------------
__global__ void __launch_bounds__(256) ap_merge(
    const float* __restrict__ Tpart,
    const float* __restrict__ Mpart,
    const float* __restrict__ Spart,
    float* __restrict__ tpool)          // (32,12,768)
{
    const int b = blockIdx.x / HEADS;
    const int h = blockIdx.x % HEADS;

    float mj[SLICES], cj[SLICES];
    float m = -1e30f;
    for (int j = 0; j < SLICES; ++j) {
        mj[j] = Mpart[((size_t)b * SLICES + j) * HEADS + h];
        m = fmaxf(m, mj[j]);
    }
    float denom = 0.f;
    for (int j = 0; j < SLICES; ++j) {
        cj[j] = __expf(mj[j] - m);
        denom += cj[j] * Spart[((size_t)b * SLICES + j) * HEADS + h];
    }
    const float inv = 1.0f / denom;

    for (int d = threadIdx.x; d < DIM; d += 256) {
        float acc = 0.f;
        for (int j = 0; j < SLICES; ++j)
            acc += cj[j] * Tpart[(((size_t)b * SLICES + j) * HEADS + h) * DIM + d];
        tpool[((size_t)b * HEADS + h) * DIM + d] = acc * inv;
    }
}

// ---------------------------------------------------------------------------
// Kernel 4: ctx[b, h*64+i] = Wv[h*64+i,:] . tpool[b,h,:] + bv[h*64+i]
// ---------------------------------------------------------------------------
__global__ void __launch_bounds__(64) ap_ctx(
    const float* __restrict__ in_w,
    const float* __restrict__ in_b,
    const float* __restrict__ tpool,
    float* __restrict__ ctx)           // (32,768)
{
    const int b = blockIdx.x / HEADS;
    const int h = blockIdx.x % HEADS;
    const int i = threadIdx.x;
    const int row = 2 * DIM + h * DH + i;
    const float* w  = in_w + (size_t)row * DIM;
    const float* tp = tpool + ((size_t)b * HEADS + h) * DIM;
    float acc = in_b[row];
    for (int d = 0; d < DIM; ++d) acc += w[d] * tp[d];
    ctx[(size_t)b * DIM + h * DH + i] = acc;
}

// ---------------------------------------------------------------------------
// Kernel 5: out[b,:] = ctx[b,:] @ out_w.T + out_b
// ---------------------------------------------------------------------------
__global__ void __launch_bounds__(256) ap_out(
    const float* __restrict__ out_w,
    const float* __restrict__ out_b,
    const float* __restrict__ ctx,
    float* __restrict__ out)           // (32,768)
{
    __shared__ float c[DIM];
    const int b = blockIdx.x;
    for (int d = threadIdx.x; d < DIM; d += 256) c[d] = ctx[(size_t)b * DIM + d];
    __syncthreads();
    for (int j = threadIdx.x; j < DIM; j += 256) {
        const float* w = out_w + (size_t)j * DIM;
        float acc = out_b[j];
        for (int d = 0; d < DIM; ++d) acc += w[d] * c[d];
        out[(size_t)b * DIM + j] = acc;
    }
}

// ---------------------------------------------------------------------------
extern "C" void kernel_launch(void* const* d_in, const int* in_sizes, int n_in,
                              void* d_out, int out_size, void* d_ws, size_t ws_size,
                              hipStream_t stream) {
    const float* tokens = (const float*)d_in[0];
    const float* query  = (const float*)d_in[1];
    const float* in_w   = (const float*)d_in[2];
    const float* in_b   = (const float*)d_in[3];
    const float* out_w  = (const float*)d_in[4];
    const float* out_b  = (const float*)d_in[5];
    float* out = (float*)d_out;

    float* ws    = (float*)d_ws;
    float* Rfrag = ws;                                            // 12288 floats
    float* cb    = ws + 12288;                                    // 16
    float* Tpart = ws + 12304;                                    // 32*16*12*768
    float* Mpart = Tpart + (size_t)BATCH * SLICES * HEADS * DIM;  // 32*16*12
    float* Spart = Mpart + (size_t)BATCH * SLICES * HEADS;        // 32*16*12
    float* tpool = Spart + (size_t)BATCH * SLICES * HEADS;        // 32*12*768
    float* ctx   = tpool + (size_t)BATCH * HEADS * DIM;           // 32*768

    ap_prep <<<1,   256, 0, stream>>>(query, in_w, in_b, Rfrag, cb);
    ap_flash<<<BATCH * SLICES, 512, 0, stream>>>(tokens, Rfrag, cb, Tpart, Mpart, Spart);
    ap_merge<<<BATCH * HEADS, 256, 0, stream>>>(Tpart, Mpart, Spart, tpool);
    ap_ctx  <<<BATCH * HEADS, 64, 0, stream>>>(in_w, in_b, tpool, ctx);
    ap_out  <<<BATCH, 256, 0, stream>>>(out_w, out_b, ctx, out);
}